// HouseholderLayer_11501922419255
// MI455X (gfx1250) — compile-verified
//
#include <hip/hip_runtime.h>

// CDNA5 / gfx1250. Compact-WY Householder apply:
//   G = V V^T and w = V x via v_wmma_f32_16x16x4_f32 (fp32 WMMA),
//   scalar backward recurrence for a, then y = x - V^T a.
// One wave32 per (n_in, n_out) batch. Bandwidth-bound: ~34 MB DRAM traffic.

typedef __attribute__((ext_vector_type(2))) float v2f;
typedef __attribute__((ext_vector_type(8))) float v8f;

#define N_IN   16
#define N_OUT  64
#define MREF   32
#define SDIM   256
#define WPB    8      // waves per block
#define GPITCH 33     // LDS pitch for 32x32 G (odd -> conflict-free column reads)

__global__ __launch_bounds__(256) void householder_wy_kernel(
    const float* __restrict__ x,   // [n_in, s, n_out]
    const float* __restrict__ v,   // [n_in, n_out, m, s]
    float* __restrict__ out)       // [n_in, s, n_out]
{
    const int wave = threadIdx.x >> 5;
    const int lane = threadIdx.x & 31;
    const int half = lane >> 4;       // which 16-lane half
    const int lr   = lane & 15;

    const int batch = blockIdx.x * WPB + wave;      // 0..1023
    const int i_in  = batch >> 6;
    const int o     = batch & 63;

    const float* vb = v + (size_t)batch * (MREF * SDIM);           // V: 32 x 256, row-major
    const float* xb = x + (size_t)(i_in * SDIM) * N_OUT + o;       // x[k] = xb[k*N_OUT]

    __shared__ float sG[WPB][MREF * GPITCH];
    __shared__ float sW[WPB][MREF];
    __shared__ float sA[WPB][MREF];
    __shared__ float sX[WPB][SDIM];

    // ---------------- Phase 0: stage this batch's x column into LDS (read x once) ----------
    float xr[8];
#pragma unroll
    for (int t = 0; t < 8; ++t) {
        xr[t] = xb[(size_t)(lane + 32 * t) * N_OUT];
        sX[wave][lane + 32 * t] = xr[t];
    }
    __syncthreads();   // uniform; orders LDS writes before cross-lane reads below

    // ---------------- Phase 1: G = V V^T (upper tiles) and w = V x via WMMA ----------------
    // f32 16x16x4 A layout: element (m,k) -> vgpr (k&1), lane m + 16*(k>>1).
    // B (4x16) layout is the register-transpose of A, so V-row-block regs serve as both.
    v8f c00 = {}; v8f c01 = {}; v8f c11 = {}; v8f cw0 = {}; v8f cw1 = {};

#pragma unroll 4
    for (int k = 0; k < SDIM; k += 4) {
        const int kk = k + 2 * half;                      // this half covers K = kk, kk+1
        v2f a0 = *(const v2f*)(vb + (size_t)lr * SDIM + kk);         // rows 0..15
        v2f a1 = *(const v2f*)(vb + (size_t)(16 + lr) * SDIM + kk);  // rows 16..31
        const float x0 = sX[wave][kk];                    // LDS broadcast per half
        const float x1 = sX[wave][kk + 1];
        v2f b2;                                           // x as B-matrix column N=0
        b2.x = (lr == 0) ? x0 : 0.0f;                     // branch-free v_cndmask
        b2.y = (lr == 0) ? x1 : 0.0f;
        c00 = __builtin_amdgcn_wmma_f32_16x16x4_f32(false, a0, false, a0, (short)0, c00, false, false);
        c01 = __builtin_amdgcn_wmma_f32_16x16x4_f32(false, a0, false, a1, (short)0, c01, false, false);
        c11 = __builtin_amdgcn_wmma_f32_16x16x4_f32(false, a1, false, a1, (short)0, c11, false, false);
        cw0 = __builtin_amdgcn_wmma_f32_16x16x4_f32(false, a0, false, b2, (short)0, cw0, false, false);
        cw1 = __builtin_amdgcn_wmma_f32_16x16x4_f32(false, a1, false, b2, (short)0, cw1, false, false);
    }

    // C layout: vgpr j holds (M = j + 8*half, N = lr). Scatter upper-triangular part of G + w.
#pragma unroll
    for (int j = 0; j < 8; ++j) {
        const int row = j + 8 * half;
        sG[wave][row * GPITCH + lr]               = c00[j];   // G[0..15][0..15]
        sG[wave][row * GPITCH + 16 + lr]          = c01[j];   // G[0..15][16..31]
        sG[wave][(16 + row) * GPITCH + 16 + lr]   = c11[j];   // G[16..31][16..31]
        if (lr == 0) {                                        // w lives in column N=0
            sW[wave][row]      = cw0[j];
            sW[wave][16 + row] = cw1[j];
        }
    }
    __syncthreads();

    // ---------------- Phase 2: backward recurrence a_i = (2/G_ii)(w_i - sum_{k>i} a_k G_ik) ----
    {
        float r = sW[wave][lane];                 // lane j tracks w_j minus accumulated terms
        for (int i = MREF - 1; i >= 0; --i) {
            const float ri  = __shfl(r, i, 32);   // broadcast lane i's residual
            const float a   = 2.0f * ri / sG[wave][i * GPITCH + i];
            if (lane == i) sA[wave][i] = a;
            if (lane < i)  r -= a * sG[wave][lane * GPITCH + i];  // upper triangle, no conflicts
        }
    }
    __syncthreads();

    // ---------------- Phase 3: y = x - V^T a  (V re-read hits L2; coalesced 128B/row) --------
    float acc[8];
#pragma unroll
    for (int t = 0; t < 8; ++t)
        acc[t] = xr[t];                           // x already in registers, no global re-read

    for (int kk = 0; kk < MREF; ++kk) {
        const float ak = sA[wave][kk];
        const float* vr = vb + (size_t)kk * SDIM;
#pragma unroll
        for (int t = 0; t < 8; ++t)
            acc[t] -= ak * vr[lane + 32 * t];
    }

    float* ob = out + (size_t)(i_in * SDIM) * N_OUT + o;
#pragma unroll
    for (int t = 0; t < 8; ++t)
        ob[(size_t)(lane + 32 * t) * N_OUT] = acc[t];
}

extern "C" void kernel_launch(void* const* d_in, const int* in_sizes, int n_in,
                              void* d_out, int out_size, void* d_ws, size_t ws_size,
                              hipStream_t stream) {
    const float* x = (const float*)d_in[0];   // [16, 256, 64] f32
    const float* v = (const float*)d_in[1];   // [16, 64, 32, 256] f32
    float* out = (float*)d_out;               // [16, 256, 64] f32

    dim3 grid((N_IN * N_OUT) / WPB);          // 128 blocks
    dim3 block(WPB * 32);                     // 8 waves of 32
    householder_wy_kernel<<<grid, block, 0, stream>>>(x, v, out);
}